// AttentionBlock_31439160607266
// MI455X (gfx1250) — compile-verified
//
#include <hip/hip_runtime.h>
#include <math.h>

// ---------------------------------------------------------------------------
// AttentionBlock on MI455X (gfx1250), bandwidth-bound plan (~200MB traffic):
//   t1 = g@Wg, t2 = x[down]@Ws        -> WMMA f32 16x16x4, weights TDM'd to LDS
//   a  = (relu(bn t1)+relu(bn t2))@Wc -> wave dot + stats
//   att = sigmoid(bn a)
//   acc[j,k] += att[pairs_in]          (scalar scatter; rank-1 factorization,
//                                       K padded 27->32 => 64x less atomic traffic)
//   out = x * (acc @ W_inv_pad + b_inv) -> WMMA f32 16x16x4 fused gate
// ---------------------------------------------------------------------------

typedef __attribute__((ext_vector_type(2))) float v2f;
typedef __attribute__((ext_vector_type(8))) float v8f;

#define EPSBN 1e-5f

// workspace float offsets
#define WS_SUM_G   0      // [64]
#define WS_SS_G    64     // [64]
#define WS_SUM_S   128    // [64]
#define WS_SS_S    192    // [64]
#define WS_SUM_A   256    // [1]
#define WS_SS_A    257    // [1]
#define WS_SCALE_G 320    // [64]
#define WS_BIAS_G  384    // [64]
#define WS_SCALE_S 448    // [64]
#define WS_BIAS_S  512    // [64]
#define WS_SCALE_A 576
#define WS_BIAS_A  577
#define WS_WPAD    1024      // [32*64]
#define WS_APRE    4096      // [Ng]
#define WS_ATT     65536     // [Ng]
#define WS_T1      262144    // [Ng*64]
#define WS_T2      4194304   // [Ng*64]
#define WS_ACC     262144    // [Nx*32]  (reuses T1/T2 after they are dead)

#if __has_builtin(__builtin_amdgcn_tensor_load_to_lds)
#define AB_USE_TDM 1
typedef __attribute__((ext_vector_type(4))) unsigned int u32x4;
typedef __attribute__((ext_vector_type(8))) int          i32x8;
typedef __attribute__((ext_vector_type(4))) int          i32x4;

// 2D row-major f32 tensor load: rows x cols, contiguous (stride = cols).
// D# per CDNA5 ISA ch.8: group0 = {count, lds_addr, global_addr, type=2},
// group1 = {data_size=4B, tensor_dim0/1, tile_dim0/1, dim0_stride}.
__device__ __forceinline__ void ab_tdm_load_2d(const float* gptr,
                                               unsigned lds_byte_off,
                                               unsigned cols, unsigned rows) {
    unsigned long long ga = (unsigned long long)(size_t)gptr;
    u32x4 g0;
    g0[0] = 1u;                                        // count=1, no gather
    g0[1] = lds_byte_off;                              // lds_addr [63:32]
    g0[2] = (unsigned)(ga & 0xffffffffu);              // global_addr lo
    g0[3] = (unsigned)((ga >> 32) & 0x01ffffffu)       // global_addr hi
            | 0x80000000u;                             // type=2 ("image")
    i32x8 g1;
    g1[0] = 0x00020000;                                // data_size=2 -> 4 bytes
    g1[1] = (int)(cols << 16);                         // tensor_dim0 [15:0]
    g1[2] = (int)(rows << 16);                         // tensor_dim1 [15:0]
    g1[3] = (int)(cols << 16);                         // tile_dim0
    g1[4] = (int)rows;                                 // tile_dim1
    g1[5] = (int)cols;                                 // tensor_dim0_stride lo
    g1[6] = 0; g1[7] = 0;
    i32x4 z4 = {0, 0, 0, 0};
    i32x8 z8 = {0, 0, 0, 0, 0, 0, 0, 0};
    __builtin_amdgcn_tensor_load_to_lds(g0, g1, z4, z4, z8, 0);
}
#endif

__global__ void ab_init_stats(float* ws) {
    int i = blockIdx.x * blockDim.x + threadIdx.x;
    if (i < 1024) ws[i] = 0.0f;
}

// ---- fused branch GEMMs: t1 = g@Wg (K=128), t2 = x[down_idx]@Ws (K=64) ----
// block = 128 threads = 4 waves; wave w owns output columns [16w, 16w+16)
__global__ void ab_branch_gemm(const float* __restrict__ g,
                               const float* __restrict__ x,
                               const int*   __restrict__ down_idx,
                               const float* __restrict__ Wg,
                               const float* __restrict__ Ws,
                               float* __restrict__ ws, int Ng) {
    __shared__ float ldsW[(128 + 64) * 64];   // Wg [128x64] then Ws [64x64]

#if AB_USE_TDM
    if (threadIdx.x < 32) {                    // one wave issues both DMAs
        unsigned base = (unsigned)(size_t)&ldsW[0];
        ab_tdm_load_2d(Wg, base,                   64u, 128u);
        ab_tdm_load_2d(Ws, base + 128u * 64u * 4u, 64u, 64u);
        __builtin_amdgcn_s_wait_tensorcnt(0);
    }
#else
    for (int i = threadIdx.x; i < (128 + 64) * 64; i += blockDim.x)
        ldsW[i] = (i < 128 * 64) ? Wg[i] : Ws[i - 128 * 64];
#endif
    __syncthreads();

    const int lane  = threadIdx.x & 31;
    const int wave  = threadIdx.x >> 5;
    const int n0    = wave << 4;
    const int mrow  = lane & 15;
    const int khalf = (lane >> 4) << 1;   // 0 or 2
    const int ncol  = lane & 15;
    const int rbase = (lane >> 4) << 3;   // C-frag row offset 0 or 8

    // B fragment base in LDS: row (khalf), column (n0+ncol); k0 steps become
    // immediate DS offsets (stride 64 floats per k row).
    const float* wBg = ldsW + khalf * 64 + n0 + ncol;
    const float* wBs = ldsW + 128 * 64 + khalf * 64 + n0 + ncol;

    float* t1 = ws + WS_T1;
    float* t2 = ws + WS_T2;

    float s1 = 0.f, q1 = 0.f, s2 = 0.f, q2 = 0.f;
    const int nStrips = (Ng + 15) >> 4;

    for (int strip = blockIdx.x; strip < nStrips; strip += gridDim.x) {
        const int row0 = strip << 4;
        const int arow = min(row0 + mrow, Ng - 1);
        {
            int nxt = strip + gridDim.x;
            if (nxt < nStrips)
                __builtin_prefetch(g + (size_t)((nxt << 4) + mrow) * 128, 0, 1);
        }

        // ---------- t1 = g @ Wg, K = 128 ----------
        v8f c1 = {};
        const float* Ap = g + (size_t)arow * 128 + khalf;   // 8B aligned
        #pragma unroll 4
        for (int k0 = 0; k0 < 128; k0 += 4) {
            float2 av = *reinterpret_cast<const float2*>(Ap + k0); // imm offset
            v2f a, b;
            a.x = av.x; a.y = av.y;
            b.x = wBg[k0 * 64];
            b.y = wBg[k0 * 64 + 64];
            c1 = __builtin_amdgcn_wmma_f32_16x16x4_f32(false, a, false, b,
                                                       (short)0, c1, false, false);
        }

        // ---------- t2 = x[down_idx] @ Ws, K = 64 ----------
        v8f c2 = {};
        const int src = down_idx[arow];
        const float* Xp = x + (size_t)src * 64 + khalf;
        #pragma unroll 4
        for (int k0 = 0; k0 < 64; k0 += 4) {
            float2 av = *reinterpret_cast<const float2*>(Xp + k0);
            v2f a, b;
            a.x = av.x; a.y = av.y;
            b.x = wBs[k0 * 64];
            b.y = wBs[k0 * 64 + 64];
            c2 = __builtin_amdgcn_wmma_f32_16x16x4_f32(false, a, false, b,
                                                       (short)0, c2, false, false);
        }

        // ---------- store + stats (unguarded fast path for full strips) ----
        if (row0 + 16 <= Ng) {
            float* p1 = t1 + (size_t)(row0 + rbase) * 64 + n0 + ncol;
            float* p2 = t2 + (size_t)(row0 + rbase) * 64 + n0 + ncol;
            #pragma unroll
            for (int r = 0; r < 8; ++r) {
                float v = c1[r];
                p1[r * 64] = v;  s1 += v;  q1 += v * v;
                float u = c2[r];
                p2[r * 64] = u;  s2 += u;  q2 += u * u;
            }
        } else {
            #pragma unroll
            for (int r = 0; r < 8; ++r) {
                int m = row0 + rbase + r;
                if (m < Ng) {
                    float v = c1[r];
                    t1[(size_t)m * 64 + n0 + ncol] = v;  s1 += v;  q1 += v * v;
                    float u = c2[r];
                    t2[(size_t)m * 64 + n0 + ncol] = u;  s2 += u;  q2 += u * u;
                }
            }
        }
    }

    // lanes l and l^16 cover the same output channel (n0+ncol)
    s1 += __shfl_xor(s1, 16, 32);  q1 += __shfl_xor(q1, 16, 32);
    s2 += __shfl_xor(s2, 16, 32);  q2 += __shfl_xor(q2, 16, 32);
    if (lane < 16) {
        int c = n0 + ncol;
        atomicAdd(&ws[WS_SUM_G + c], s1);
        atomicAdd(&ws[WS_SS_G  + c], q1);
        atomicAdd(&ws[WS_SUM_S + c], s2);
        atomicAdd(&ws[WS_SS_S  + c], q2);
    }
}

// ---- fold BN stats into per-channel scale/bias; pad W_inv 27x64 -> 32x64 ----
__global__ void ab_finalize_stats(float* ws,
                                  const float* __restrict__ gamma_g,
                                  const float* __restrict__ beta_g,
                                  const float* __restrict__ gamma_s,
                                  const float* __restrict__ beta_s,
                                  const float* __restrict__ W_inv, int Ng) {
    int c = threadIdx.x;            // 64 threads
    float invN = 1.0f / (float)Ng;
    float mg = ws[WS_SUM_G + c] * invN;
    float vg = fmaxf(ws[WS_SS_G + c] * invN - mg * mg, 0.f);
    float ag = gamma_g[c] * rsqrtf(vg + EPSBN);
    ws[WS_SCALE_G + c] = ag;
    ws[WS_BIAS_G  + c] = beta_g[c] - mg * ag;

    float ms = ws[WS_SUM_S + c] * invN;
    float vs = fmaxf(ws[WS_SS_S + c] * invN - ms * ms, 0.f);
    float as = gamma_s[c] * rsqrtf(vs + EPSBN);
    ws[WS_SCALE_S + c] = as;
    ws[WS_BIAS_S  + c] = beta_s[c] - ms * as;

    float* wp = ws + WS_WPAD;
    for (int k = 0; k < 32; ++k)
        wp[k * 64 + c] = (k < 27) ? W_inv[k * 64 + c] : 0.0f;
}

// ---- a_pre[i] = (relu(bn(t1)) + relu(bn(t2))) . Wc, accumulate stats ----
__global__ void ab_attention_pre(float* ws, const float* __restrict__ Wc, int Ng) {
    const int lane = threadIdx.x & 31;
    const int wave = threadIdx.x >> 5;
    const int wpg  = (blockDim.x >> 5) * gridDim.x;
    int gw = blockIdx.x * (blockDim.x >> 5) + wave;

    const float* t1 = ws + WS_T1;
    const float* t2 = ws + WS_T2;
    const int c0 = lane, c1 = lane + 32;
    const float sg0 = ws[WS_SCALE_G + c0], bg0 = ws[WS_BIAS_G + c0];
    const float sg1 = ws[WS_SCALE_G + c1], bg1 = ws[WS_BIAS_G + c1];
    const float ss0 = ws[WS_SCALE_S + c0], bs0 = ws[WS_BIAS_S + c0];
    const float ss1 = ws[WS_SCALE_S + c1], bs1 = ws[WS_BIAS_S + c1];
    const float wc0 = Wc[c0], wc1 = Wc[c1];

    float sa = 0.f, qa = 0.f;
    for (int i = gw; i < Ng; i += wpg) {
        size_t base = (size_t)i * 64;
        float h0 = fmaxf(t1[base + c0] * sg0 + bg0, 0.f) +
                   fmaxf(t2[base + c0] * ss0 + bs0, 0.f);
        float h1 = fmaxf(t1[base + c1] * sg1 + bg1, 0.f) +
                   fmaxf(t2[base + c1] * ss1 + bs1, 0.f);
        h0 = fmaxf(h0, 0.f); h1 = fmaxf(h1, 0.f);   // relu(phi+theta)
        float part = h0 * wc0 + h1 * wc1;
        #pragma unroll
        for (int off = 16; off > 0; off >>= 1) part += __shfl_xor(part, off, 32);
        if (lane == 0) {
            ws[WS_APRE + i] = part;
            sa += part; qa += part * part;
        }
    }
    if (lane == 0) {
        atomicAdd(&ws[WS_SUM_A], sa);
        atomicAdd(&ws[WS_SS_A],  qa);
    }
}

__global__ void ab_finalize_a(float* ws, const float* __restrict__ gamma_c,
                              const float* __restrict__ beta_c, int Ng) {
    float invN = 1.0f / (float)Ng;
    float m = ws[WS_SUM_A] * invN;
    float v = fmaxf(ws[WS_SS_A] * invN - m * m, 0.f);
    float a = gamma_c[0] * rsqrtf(v + EPSBN);
    ws[WS_SCALE_A] = a;
    ws[WS_BIAS_A]  = beta_c[0] - m * a;
}

__global__ void ab_make_att(float* ws, int Ng) {
    float sa = ws[WS_SCALE_A], ba = ws[WS_BIAS_A];
    for (int i = blockIdx.x * blockDim.x + threadIdx.x; i < Ng;
         i += blockDim.x * gridDim.x) {
        float z = ws[WS_APRE + i] * sa + ba;
        ws[WS_ATT + i] = 1.0f / (1.0f + expf(-z));
    }
}

__global__ void ab_zero_acc(float4* __restrict__ acc4, int n4) {
    for (int i = blockIdx.x * blockDim.x + threadIdx.x; i < n4;
         i += blockDim.x * gridDim.x)
        acc4[i] = make_float4(0.f, 0.f, 0.f, 0.f);
}

// ---- scalar scatter: acc[pairs_out, k] += att[pairs_in] ----
__global__ void ab_scatter(const int* __restrict__ pairs_in,
                           const int* __restrict__ pairs_out,
                           float* ws, int P) {
    const int k = blockIdx.y;                        // 0..26
    const float* att = ws + WS_ATT;
    float* acc = ws + WS_ACC;
    const size_t rowb = (size_t)k * P;
    for (int p = blockIdx.x * blockDim.x + threadIdx.x; p < P;
         p += blockDim.x * gridDim.x) {
        int i = pairs_in [rowb + p];
        int j = pairs_out[rowb + p];
        atomicAdd(&acc[(size_t)j * 32 + k], att[i]);
    }
}

// ---- up = acc @ W_inv_pad (WMMA, K=32), out = x * (up + b_inv) ----
__global__ void ab_upconv_gate(const float* __restrict__ x,
                               const float* __restrict__ b_inv,
                               const float* __restrict__ ws,
                               float* __restrict__ out, int Nx) {
    const int lane  = threadIdx.x & 31;
    const int wave  = threadIdx.x >> 5;
    const int n0    = wave << 4;
    const int mrow  = lane & 15;
    const int khalf = (lane >> 4) << 1;
    const int ncol  = lane & 15;
    const int rbase = (lane >> 4) << 3;

    const float* acc = ws + WS_ACC;
    const float* wp  = ws + WS_WPAD;

    // preload all B fragments (W_inv_pad is 8 KB, L2 resident)
    v2f bf[8];
    #pragma unroll
    for (int kk = 0; kk < 8; ++kk) {
        int k0 = (kk << 2) + khalf;
        bf[kk].x = wp[(size_t)k0 * 64 + n0 + ncol];
        bf[kk].y = wp[(size_t)(k0 + 1) * 64 + n0 + ncol];
    }
    const float bias = b_inv[n0 + ncol];

    const int nStrips = (Nx + 15) >> 4;
    for (int strip = blockIdx.x; strip < nStrips; strip += gridDim.x) {
        const int row0 = strip << 4;
        const int arow = min(row0 + mrow, Nx - 1);
        {
            int nxt = strip + gridDim.x;
            if (nxt < nStrips)
                __builtin_prefetch(acc + (size_t)((nxt << 4) + mrow) * 32, 0, 1);
        }
        const float* Ap = acc + (size_t)arow * 32 + khalf;
        v8f c = {};
        #pragma unroll
        for (int kk = 0; kk < 8; ++kk) {
            float2 av = *reinterpret_cast<const float2*>(Ap + (kk << 2));
            v2f a; a.x = av.x; a.y = av.y;
            c = __builtin_amdgcn_wmma_f32_16x16x4_f32(false, a, false, bf[kk],
                                                      (short)0, c, false, false);
        }
        if (row0 + 16 <= Nx) {
            const float* xp = x   + (size_t)(row0 + rbase) * 64 + n0 + ncol;
            float*       op = out + (size_t)(row0 + rbase) * 64 + n0 + ncol;
            #pragma unroll
            for (int r = 0; r < 8; ++r)
                op[r * 64] = xp[r * 64] * (c[r] + bias);
        } else {
            #pragma unroll
            for (int r = 0; r < 8; ++r) {
                int m = row0 + rbase + r;
                if (m < Nx) {
                    size_t idx = (size_t)m * 64 + n0 + ncol;
                    out[idx] = x[idx] * (c[r] + bias);
                }
            }
        }
    }
}

extern "C" void kernel_launch(void* const* d_in, const int* in_sizes, int n_in,
                              void* d_out, int out_size, void* d_ws, size_t ws_size,
                              hipStream_t stream) {
    const float* g         = (const float*)d_in[0];
    const float* x         = (const float*)d_in[1];
    const int*   down_idx  = (const int*)  d_in[2];
    const int*   pairs_in  = (const int*)  d_in[3];
    const int*   pairs_out = (const int*)  d_in[4];
    const float* Wg        = (const float*)d_in[5];
    const float* Ws        = (const float*)d_in[6];
    const float* Wc        = (const float*)d_in[7];
    const float* W_inv     = (const float*)d_in[8];
    const float* b_inv     = (const float*)d_in[9];
    const float* gamma_g   = (const float*)d_in[10];
    const float* beta_g    = (const float*)d_in[11];
    const float* gamma_s   = (const float*)d_in[12];
    const float* beta_s    = (const float*)d_in[13];
    const float* gamma_c   = (const float*)d_in[14];
    const float* beta_c    = (const float*)d_in[15];
    float* out = (float*)d_out;
    float* ws  = (float*)d_ws;

    const int Ng = in_sizes[2];            // down_idx length
    const int Nx = in_sizes[1] / 64;       // x rows
    const int K  = 27;
    const int P  = in_sizes[3] / K;

    ab_init_stats<<<4, 256, 0, stream>>>(ws);

    int nStripsG = (Ng + 15) >> 4;
    int gemmBlocks = nStripsG < 960 ? nStripsG : 960;
    ab_branch_gemm<<<gemmBlocks, 128, 0, stream>>>(g, x, down_idx, Wg, Ws, ws, Ng);

    ab_finalize_stats<<<1, 64, 0, stream>>>(ws, gamma_g, beta_g, gamma_s, beta_s,
                                            W_inv, Ng);

    ab_attention_pre<<<256, 256, 0, stream>>>(ws, Wc, Ng);

    ab_finalize_a<<<1, 1, 0, stream>>>(ws, gamma_c, beta_c, Ng);
    ab_make_att<<<(Ng + 255) / 256, 256, 0, stream>>>(ws, Ng);

    int n4 = (Nx * 32) / 4;
    ab_zero_acc<<<2048, 256, 0, stream>>>((float4*)(ws + WS_ACC), n4);

    dim3 sgrid((P + 255) / 256 < 400 ? (P + 255) / 256 : 400, K);
    ab_scatter<<<sgrid, 256, 0, stream>>>(pairs_in, pairs_out, ws, P);

    int nStripsX = (Nx + 15) >> 4;
    int upBlocks = nStripsX < 1920 ? nStripsX : 1920;
    ab_upconv_gate<<<upBlocks, 128, 0, stream>>>(x, b_inv, ws, out, Nx);
}